// Voxelization_19739669693074
// MI455X (gfx1250) — compile-verified
//
#include <hip/hip_runtime.h>
#include <hip/hip_bf16.h>
#include <stdint.h>

#define MV 160000   // max_voxels (module constant)
#define MP 10       // max_points (module constant)
#define CF 5        // point feature count

// ---------- fused init: zero outputs, reset hash + per-voxel arrays ----------
__global__ void k_init(int* __restrict__ out, int zn,
                       int* __restrict__ hkey, int* __restrict__ hval, int H,
                       int* __restrict__ counts, int* __restrict__ fillctr) {
  int i = blockIdx.x * blockDim.x + threadIdx.x;
  if (i < zn) out[i] = 0;                         // voxels + coors regions
  if (i < H) { hkey[i] = -1; hval[i] = 0x7fffffff; }
  if (i < MV) { counts[i] = 0; fillctr[i] = 0; }
}

// ---------- phase 1: quantize + hash insert + per-voxel min point index ----------
__global__ void k_assign(const float* __restrict__ pts, const float* __restrict__ vsz,
                         const float* __restrict__ crange, int N,
                         int* __restrict__ hkey, int* __restrict__ hval,
                         int* __restrict__ pslot, unsigned HM) {
  int i = blockIdx.x * blockDim.x + threadIdx.x;
  if (i >= N) return;
  float vx = vsz[0], vy = vsz[1], vz = vsz[2];
  float mx = crange[0], my = crange[1], mz = crange[2];
  int gx = __float2int_rn((crange[3] - mx) / vx);
  int gy = __float2int_rn((crange[4] - my) / vy);
  int gz = __float2int_rn((crange[5] - mz) / vz);
  float px = pts[(size_t)i * CF + 0];
  float py = pts[(size_t)i * CF + 1];
  float pz = pts[(size_t)i * CF + 2];
  int cx = (int)floorf((px - mx) / vx);
  int cy = (int)floorf((py - my) / vy);
  int cz = (int)floorf((pz - mz) / vz);
  bool valid = (cx >= 0) && (cx < gx) && (cy >= 0) && (cy < gy) && (cz >= 0) && (cz < gz);
  if (!valid) { pslot[i] = -1; return; }
  int lin = (cx * gy + cy) * gz + cz;             // < 82.9M, fits int32
  unsigned slot = ((unsigned)lin * 2654435761u) & HM;
  for (;;) {
    int k = atomicCAS(&hkey[slot], -1, lin);      // claim if empty
    if (k == -1 || k == lin) break;               // per-key chain converges to one slot
    slot = (slot + 1) & HM;
  }
  atomicMin(&hval[slot], i);                      // first-appearance point index
  pslot[i] = (int)slot;
}

// ---------- phase 2a: head flags + wave32 ballot block scan ----------
__global__ void k_heads(const int* __restrict__ pslot, const int* __restrict__ hval,
                        int N, int* __restrict__ flags, int* __restrict__ ranks,
                        int* __restrict__ bsums) {
  int i = blockIdx.x * 256 + threadIdx.x;
  int f = 0;
  if (i < N) { int s = pslot[i]; if (s >= 0 && hval[s] == i) f = 1; }
  unsigned mask = (unsigned)__ballot(f);          // wave32: low 32 bits
  int lane = threadIdx.x & 31;
  int wid  = threadIdx.x >> 5;
  __shared__ int wsum[8];
  if (lane == 0) wsum[wid] = __popc(mask);
  __syncthreads();
  int woff = 0;
  for (int w = 0; w < wid; ++w) woff += wsum[w];
  if (i < N) {
    flags[i] = f;
    ranks[i] = woff + __popc(mask & ((1u << lane) - 1u));  // block-local exclusive
  }
  if (threadIdx.x == 0) {
    int t = 0;
    for (int w = 0; w < 8; ++w) t += wsum[w];
    bsums[blockIdx.x] = t;
  }
}

// ---------- generic block scan (shfl + LDS), values not just 0/1 ----------
__global__ void k_blockscan(const int* __restrict__ in, int n,
                            int* __restrict__ localExcl, int* __restrict__ bsums) {
  int i = blockIdx.x * 256 + threadIdx.x;
  int v = (i < n) ? in[i] : 0;
  int lane = threadIdx.x & 31;
  int wid  = threadIdx.x >> 5;
  int x = v;
  for (int d = 1; d < 32; d <<= 1) {
    int y = __shfl_up(x, d, 32);
    if (lane >= d) x += y;
  }
  __shared__ int wsum[8];
  if (lane == 31) wsum[wid] = x;
  __syncthreads();
  int woff = 0;
  for (int w = 0; w < wid; ++w) woff += wsum[w];
  if (i < n) localExcl[i] = woff + (x - v);
  if (threadIdx.x == 0) {
    int t = 0;
    for (int w = 0; w < 8; ++w) t += wsum[w];
    bsums[blockIdx.x] = t;
  }
}

// ---------- single-workgroup chained scan of block sums ----------
__global__ void k_scan_bsums(int* __restrict__ bsums, int nb, int* __restrict__ total) {
  __shared__ int wsum[8];
  __shared__ int carry;
  int tid = threadIdx.x;
  int lane = tid & 31, wid = tid >> 5;
  if (tid == 0) carry = 0;
  __syncthreads();
  for (int base = 0; base < nb; base += 256) {
    int idx = base + tid;
    int v = (idx < nb) ? bsums[idx] : 0;
    int x = v;
    for (int d = 1; d < 32; d <<= 1) {
      int y = __shfl_up(x, d, 32);
      if (lane >= d) x += y;
    }
    if (lane == 31) wsum[wid] = x;
    __syncthreads();
    int woff = 0;
    for (int w = 0; w < wid; ++w) woff += wsum[w];
    int excl = carry + woff + (x - v);
    if (idx < nb) bsums[idx] = excl;
    __syncthreads();
    if (tid == 0) { int t = 0; for (int w = 0; w < 8; ++w) t += wsum[w]; carry += t; }
    __syncthreads();
  }
  if (tid == 0) *total = carry;
}

// ---------- phase 2c: heads publish voxel id into hash slot, write coors ----------
__global__ void k_vid_heads(const float* __restrict__ pts, const float* __restrict__ vsz,
                            const float* __restrict__ crange, int N,
                            const int* __restrict__ flags, const int* __restrict__ ranks,
                            const int* __restrict__ bsums, const int* __restrict__ pslot,
                            int* __restrict__ hval, int* __restrict__ outCoors) {
  int i = blockIdx.x * blockDim.x + threadIdx.x;
  if (i >= N || !flags[i]) return;
  int r = ranks[i] + bsums[i >> 8];               // global appearance rank
  int s = pslot[i];
  if (r < MV) {
    hval[s] = r;                                  // written exactly once (by its head)
    float vx = vsz[0], vy = vsz[1], vz = vsz[2];
    float mx = crange[0], my = crange[1], mz = crange[2];
    int cx = (int)floorf((pts[(size_t)i * CF + 0] - mx) / vx);
    int cy = (int)floorf((pts[(size_t)i * CF + 1] - my) / vy);
    int cz = (int)floorf((pts[(size_t)i * CF + 2] - mz) / vz);
    outCoors[(size_t)r * 3 + 0] = cz;             // reversed (z, y, x)
    outCoors[(size_t)r * 3 + 1] = cy;
    outCoors[(size_t)r * 3 + 2] = cx;
  } else {
    hval[s] = MV;                                 // dropped-voxel sentinel
  }
}

// ---------- phase 3: per-point voxel id + uncapped per-voxel counts ----------
__global__ void k_vid_points(const int* __restrict__ pslot, const int* __restrict__ hval,
                             int N, int* __restrict__ vid, int* __restrict__ counts) {
  int i = blockIdx.x * blockDim.x + threadIdx.x;
  if (i >= N) return;
  int s = pslot[i];
  int v = (s >= 0) ? hval[s] : MV;
  vid[i] = v;
  if (v < MV) atomicAdd(&counts[v], 1);
}

// ---------- phase 4: scatter point indices into CSR lists (order canonicalized later) ----
__global__ void k_scatter(const int* __restrict__ vid, int N,
                          const int* __restrict__ offsLocal, const int* __restrict__ bsums2,
                          int* __restrict__ fillctr, int* __restrict__ plist) {
  int i = blockIdx.x * blockDim.x + threadIdx.x;
  if (i >= N) return;
  int v = vid[i];
  if (v >= MV) return;
  int base = offsLocal[v] + bsums2[v >> 8];
  int pos = atomicAdd(&fillctr[v], 1);
  plist[base + pos] = i;
}

// ---------- phase 5: per-voxel iterative-min selection of <=MP smallest indices + gather ----
__global__ void k_selgather(const float* __restrict__ pts, const int* __restrict__ counts,
                            const int* __restrict__ offsLocal, const int* __restrict__ bsums2,
                            const int* __restrict__ plist, const int* __restrict__ headcnt,
                            float* __restrict__ outVox, int* __restrict__ outNum,
                            int* __restrict__ outVoxNum) {
  int v = blockIdx.x * blockDim.x + threadIdx.x;
  if (v >= MV) return;
  int len  = counts[v];
  int kept = (len < MP) ? len : MP;
  outNum[v] = kept;
  if (v == 0) *outVoxNum = ((*headcnt) < MV) ? (*headcnt) : MV;
  if (kept == 0) return;
  int base = offsLocal[v] + bsums2[v >> 8];
  int prev = -1;
  for (int r = 0; r < kept; ++r) {
    int m = 0x7fffffff;
    for (int j = 0; j < len; ++j) {               // short list; L2-resident
      int idx = plist[base + j];
      if (idx > prev && idx < m) m = idx;
    }
    prev = m;
    float* dst = outVox + ((size_t)v * MP + r) * CF;
    const float* src = pts + (size_t)m * CF;
#pragma unroll
    for (int c = 0; c < CF; ++c) dst[c] = src[c];
  }
}

extern "C" void kernel_launch(void* const* d_in, const int* in_sizes, int n_in,
                              void* d_out, int out_size, void* d_ws, size_t ws_size,
                              hipStream_t stream) {
  const float* pts    = (const float*)d_in[0];
  const float* vsz    = (const float*)d_in[1];
  const float* crange = (const float*)d_in[2];
  (void)n_in; (void)out_size;

  int N   = in_sizes[0] / CF;                     // 1,200,000
  int NB  = (N + 255) / 256;                      // point-kernel blocks
  int NBV = (MV + 255) / 256;                     // voxel-kernel blocks (625)

  // hash: unique voxels <= N < 2^21 so 2^21 can never fill; use 2^22 if ws allows
  size_t fixedInts = (size_t)5 * N + NB + NBV + 2 + (size_t)3 * MV;
  int H = 1 << 22;
  if ((2ull * H + fixedInts) * 4ull > ws_size) H = 1 << 21;
  unsigned HM = (unsigned)(H - 1);

  int* w = (int*)d_ws;
  int* hkey      = w;  w += H;
  int* hval      = w;  w += H;
  int* pslot     = w;  w += N;
  int* vid       = w;  w += N;
  int* flags     = w;  w += N;
  int* ranks     = w;  w += N;
  int* plist     = w;  w += N;
  int* bsums     = w;  w += NB;
  int* bsums2    = w;  w += NBV;
  int* headcnt   = w;  w += 1;
  int* dummyTot  = w;  w += 1;
  int* counts    = w;  w += MV;
  int* offsLocal = w;  w += MV;
  int* fillctr   = w;  w += MV;

  float* outVox    = (float*)d_out;
  int*   outCoors  = (int*)d_out + (size_t)MV * MP * CF;
  int*   outNum    = outCoors + (size_t)MV * 3;
  int*   outVoxNum = outNum + MV;

  const int T = 256;
  int zn = MV * MP * CF + MV * 3;                 // voxels + coors must be re-zeroed each call
  int nInit = zn > H ? zn : H;

  k_init<<<(nInit + T - 1) / T, T, 0, stream>>>((int*)d_out, zn, hkey, hval, H,
                                                counts, fillctr);
  k_assign<<<NB, T, 0, stream>>>(pts, vsz, crange, N, hkey, hval, pslot, HM);
  k_heads<<<NB, 256, 0, stream>>>(pslot, hval, N, flags, ranks, bsums);
  k_scan_bsums<<<1, 256, 0, stream>>>(bsums, NB, headcnt);
  k_vid_heads<<<NB, T, 0, stream>>>(pts, vsz, crange, N, flags, ranks, bsums, pslot,
                                    hval, outCoors);
  k_vid_points<<<NB, T, 0, stream>>>(pslot, hval, N, vid, counts);
  k_blockscan<<<NBV, 256, 0, stream>>>(counts, MV, offsLocal, bsums2);
  k_scan_bsums<<<1, 256, 0, stream>>>(bsums2, NBV, dummyTot);
  k_scatter<<<NB, T, 0, stream>>>(vid, N, offsLocal, bsums2, fillctr, plist);
  k_selgather<<<NBV, 256, 0, stream>>>(pts, counts, offsLocal, bsums2, plist, headcnt,
                                       outVox, outNum, outVoxNum);
}